// GNNModel_47674136985664
// MI455X (gfx1250) — compile-verified
//
#include <hip/hip_runtime.h>
#include <hip/hip_bf16.h>

typedef __attribute__((ext_vector_type(2))) float v2f;
typedef __attribute__((ext_vector_type(8))) float v8f;

// ---------------------------------------------------------------------------
// Degree / normalization
// ---------------------------------------------------------------------------
__global__ void k_init_deg(int* __restrict__ deg, int n) {
  int i = blockIdx.x * blockDim.x + threadIdx.x;
  if (i < n) deg[i] = 1;  // self loop contributes 1
}

__global__ void k_count_deg(const int* __restrict__ dst, int* __restrict__ deg, int E) {
  int i = blockIdx.x * blockDim.x + threadIdx.x;
  if (i < E) atomicAdd(&deg[dst[i]], 1);
}

__global__ void k_dinv(const int* __restrict__ deg, float* __restrict__ dinv, int n) {
  int i = blockIdx.x * blockDim.x + threadIdx.x;
  if (i < n) dinv[i] = rsqrtf((float)deg[i]);
}

__global__ void k_zero_f32(float* __restrict__ p, long long n) {
  long long i = (long long)blockIdx.x * blockDim.x + threadIdx.x;
  if (i < n) p[i] = 0.0f;
}

// ---------------------------------------------------------------------------
// Full-precision fp32 GEMM via V_WMMA_F32_16X16X4_F32.
// One wave computes one 16x16 tile of C = A[M,K] @ W[K,N].
// A-frag (16x4 fp32, ISA 7.12.2): lanes 0-15 hold M=lane, VGPR0/1 = K=(k,k+1);
// lanes 16-31 hold K=(k+2,k+3). B-frag (4x16) mirrored: lane&15 = N column,
// lane>>4 selects K pair. C/D: VGPR v holds M = v + 8*(lane>=16), N = lane&15.
// ---------------------------------------------------------------------------
template <int K, int N>
__global__ __launch_bounds__(256) void k_gemm_wmma(const float* __restrict__ A,
                                                   const float* __restrict__ W,
                                                   float* __restrict__ C, int mTiles) {
  const int lane = threadIdx.x & 31;
  const int wave = threadIdx.x >> 5;
  const int tile = blockIdx.x * 8 + wave;
  constexpr int tilesN = N / 16;
  const int mT = tile / tilesN;
  const int nT = tile % tilesN;
  if (mT >= mTiles) return;  // wave-uniform: EXEC stays all-ones for WMMA

  const int row = lane & 15;
  const int kh = lane >> 4;  // 0 or 1
  const int m0 = mT * 16;
  const int n0 = nT * 16;
  const int col = n0 + row;

  const float* ap = A + (size_t)(m0 + row) * K + 2 * kh;

  v8f c = {};
#pragma unroll 4
  for (int k = 0; k < K; k += 4) {
    v2f a = *(const v2f*)(ap + k);  // 8B-aligned: even float offset
    const int kr = k + 2 * kh;
    v2f b;
    b.x = W[(size_t)kr * N + col];
    b.y = W[(size_t)(kr + 1) * N + col];
    c = __builtin_amdgcn_wmma_f32_16x16x4_f32(false, a, false, b, (short)0, c, false,
                                              false);
  }

  float* cp = C + (size_t)(m0 + 8 * kh) * N + col;
#pragma unroll
  for (int v = 0; v < 8; ++v) cp[(size_t)v * N] = c[v];
}

// ---------------------------------------------------------------------------
// Edge aggregation: agg[dst] += hw[src] * dinv[src]*dinv[dst]
// One thread per (edge, 4-feature chunk); float4 gather, native f32 atomics.
// ---------------------------------------------------------------------------
template <int F>
__global__ void k_edge_agg(const float* __restrict__ hw, const float* __restrict__ dinv,
                           const int* __restrict__ src, const int* __restrict__ dst,
                           float* __restrict__ agg, int E) {
  constexpr int F4 = F / 4;
  long long idx = (long long)blockIdx.x * blockDim.x + threadIdx.x;
  long long total = (long long)E * F4;
  if (idx >= total) return;
  const int e = (int)(idx / F4);
  const int q = (int)(idx % F4);
  const int s = src[e];
  const int d = dst[e];
  const float nrm = dinv[s] * dinv[d];
  const float4 m = ((const float4*)(hw + (size_t)s * F))[q];
  float* out = agg + (size_t)d * F + 4 * q;
  unsafeAtomicAdd(out + 0, m.x * nrm);
  unsafeAtomicAdd(out + 1, m.y * nrm);
  unsafeAtomicAdd(out + 2, m.z * nrm);
  unsafeAtomicAdd(out + 3, m.w * nrm);
}

// h = relu(agg + hw * dinv^2 + b); written in place over hw
template <int F>
__global__ void k_post1(float* __restrict__ hw, const float* __restrict__ agg,
                        const float* __restrict__ dinv, const float* __restrict__ b,
                        int n) {
  long long idx = (long long)blockIdx.x * blockDim.x + threadIdx.x;
  long long total = (long long)n * F;
  if (idx >= total) return;
  const int i = (int)(idx / F);
  const int f = (int)(idx % F);
  const float di = dinv[i];
  const float v = agg[idx] + hw[idx] * di * di + b[f];
  hw[idx] = v > 0.0f ? v : 0.0f;
}

// out = b + hw2 * dinv^2  (self loop + bias; edge atomics accumulate after)
template <int F>
__global__ void k_init_out(const float* __restrict__ hw2, const float* __restrict__ dinv,
                           const float* __restrict__ b, float* __restrict__ out, int n) {
  long long idx = (long long)blockIdx.x * blockDim.x + threadIdx.x;
  long long total = (long long)n * F;
  if (idx >= total) return;
  const int i = (int)(idx / F);
  const int f = (int)(idx % F);
  const float di = dinv[i];
  out[idx] = b[f] + hw2[idx] * di * di;
}

// ---------------------------------------------------------------------------
extern "C" void kernel_launch(void* const* d_in, const int* in_sizes, int n_in,
                              void* d_out, int out_size, void* d_ws, size_t ws_size,
                              hipStream_t stream) {
  (void)n_in;
  (void)out_size;
  (void)ws_size;

  const float* x = (const float*)d_in[0];
  const int* ei = (const int*)d_in[1];
  const float* W1 = (const float*)d_in[2];
  const float* b1 = (const float*)d_in[3];
  const float* W2 = (const float*)d_in[4];
  const float* b2 = (const float*)d_in[5];
  float* out = (float*)d_out;

  constexpr int IN = 256, HID = 128, OUT = 64;
  const int N = in_sizes[0] / IN;  // 100000 (multiple of 16)
  const int E = in_sizes[1] / 2;   // 1600000
  const int* src = ei;
  const int* dst = ei + E;

  // workspace partition (256B aligned)
  char* ws = (char*)d_ws;
  size_t off = 0;
  auto wsalloc = [&](size_t bytes) -> void* {
    void* p = ws + off;
    off += (bytes + 255) & ~(size_t)255;
    return p;
  };
  int* deg = (int*)wsalloc((size_t)N * 4);
  float* dinv = (float*)wsalloc((size_t)N * 4);
  float* hw1 = (float*)wsalloc((size_t)N * HID * 4);  // x@W1, then h in place
  float* agg1 = (float*)wsalloc((size_t)N * HID * 4);
  float* hw2 = (float*)wsalloc((size_t)N * OUT * 4);

  const int TB = 256;
  auto blks = [](long long total, int tb) { return (unsigned)((total + tb - 1) / tb); };

  // 1) symmetric normalization
  k_init_deg<<<blks(N, TB), TB, 0, stream>>>(deg, N);
  k_count_deg<<<blks(E, TB), TB, 0, stream>>>(dst, deg, E);
  k_dinv<<<blks(N, TB), TB, 0, stream>>>(deg, dinv, N);

  // 2) layer 1 GEMM: hw1 = x @ W1   [N,256]x[256,128]
  {
    const int mTiles = N / 16;
    const int tiles = mTiles * (HID / 16);
    k_gemm_wmma<IN, HID><<<blks(tiles, 8), TB, 0, stream>>>(x, W1, hw1, mTiles);
  }

  // 3) aggregate layer 1
  k_zero_f32<<<blks((long long)N * HID, TB), TB, 0, stream>>>(agg1, (long long)N * HID);
  k_edge_agg<HID><<<blks((long long)E * (HID / 4), TB), TB, 0, stream>>>(hw1, dinv, src,
                                                                         dst, agg1, E);
  // 4) self loop + bias + relu -> h (in place in hw1)
  k_post1<HID><<<blks((long long)N * HID, TB), TB, 0, stream>>>(hw1, agg1, dinv, b1, N);

  // 5) layer 2 GEMM: hw2 = h @ W2   [N,128]x[128,64]
  {
    const int mTiles = N / 16;
    const int tiles = mTiles * (OUT / 16);
    k_gemm_wmma<HID, OUT><<<blks(tiles, 8), TB, 0, stream>>>(hw1, W2, hw2, mTiles);
  }

  // 6) aggregate layer 2 directly into d_out
  k_init_out<OUT><<<blks((long long)N * OUT, TB), TB, 0, stream>>>(hw2, dinv, b2, out, N);
  k_edge_agg<OUT><<<blks((long long)E * (OUT / 4), TB), TB, 0, stream>>>(hw2, dinv, src,
                                                                         dst, out, E);
}